// SqueezeExcitation_34205119545903
// MI455X (gfx1250) — compile-verified
//
#include <hip/hip_runtime.h>
#include <hip/hip_bf16.h>

// ---------------------------------------------------------------------------
// Squeeze-Excitation, MI455X (gfx1250, wave32).
// B=64, C=256, H=W=56 (HW=3136), SQ=64.
//   s[64,256]   = mean_{hw} x
//   h[64,64]    = relu(s @ w1^T + b1)
//   g[64,256]   = sigmoid(h @ w2^T + b2)
//   out         = x * g (broadcast over HW)
// Bandwidth-bound: ~616 MB HBM traffic -> ~26us at 23.3 TB/s. The two dense
// layers are ~4 MFLOP and run as fp32 WMMA (V_WMMA_F32_16X16X4_F32) in a
// single 16-wave workgroup.
// ---------------------------------------------------------------------------

typedef __attribute__((ext_vector_type(2))) float v2f;
typedef __attribute__((ext_vector_type(8))) float v8f;

#define HW      3136
#define HW4     784      // HW / 4 (float4 granularity)
#define NROWS   16384    // B * C
#define CCH     256
#define SQC     64

// ---------------------------------------------------------------------------
// Kernel 1: global average pool. One wave per (b,c) row; coalesced b128 loads
// (32 lanes x 16B = 512B per issue), wave32 shuffle reduction.
// ---------------------------------------------------------------------------
__global__ void se_pool_kernel(const float* __restrict__ x, float* __restrict__ s) {
    const int wavesPerBlock = blockDim.x >> 5;
    const int row  = blockIdx.x * wavesPerBlock + (threadIdx.x >> 5);
    const int lane = threadIdx.x & 31;

    const float4* __restrict__ xr =
        reinterpret_cast<const float4*>(x) + (size_t)row * HW4;

    float sum = 0.0f;
    #pragma unroll 4
    for (int i = lane; i < HW4; i += 32) {
        float4 v = xr[i];
        sum += (v.x + v.y) + (v.z + v.w);
    }
    // wave32 butterfly reduction
    #pragma unroll
    for (int off = 16; off > 0; off >>= 1)
        sum += __shfl_xor(sum, off, 32);

    if (lane == 0) s[row] = sum * (1.0f / (float)HW);
}

// ---------------------------------------------------------------------------
// Kernel 2: excitation MLP via V_WMMA_F32_16X16X4_F32.
// Single workgroup, 512 threads = 16 waves (wave32).
//
// FC1: h[64,64] = relu(s[64,256] @ w1[64,256]^T + b1). 4x4 grid of 16x16
//      tiles, one tile per wave, K=256 in steps of 4.
// FC2: g[64,256] = sigmoid(h @ w2[256,64]^T + b2). 4x16 grid of tiles,
//      4 tiles per wave, K=64. h staged in LDS (padded stride 65).
//
// Fragment layouts per CDNA5 ISA (5_wmma.md):
//   A (16x4 f32): lane = (half<<4)|m holds A[m][2*half + {0,1}] in v[0],v[1]
//   B (4x16 f32): lane = (half<<4)|n holds B[2*half + {0,1}][n]
//   C/D (16x16 f32, 8 VGPRs): lane<16 -> N=lane, M=r; lane>=16 -> M=8+r
// EXEC stays all-ones: no divergence anywhere around the WMMA ops.
// ---------------------------------------------------------------------------
#define HSTR 65  // padded LDS stride for h (64 + 1, avoids 64-bank conflicts)

__global__ void se_excite_kernel(const float* __restrict__ s,   // [64,256]
                                 const float* __restrict__ w1,  // [64,256]
                                 const float* __restrict__ b1,  // [64]
                                 const float* __restrict__ w2,  // [256,64]
                                 const float* __restrict__ b2,  // [256]
                                 float* __restrict__ g)         // [64,256]
{
    __shared__ float h_lds[SQC * HSTR];

    const int tid   = threadIdx.x;
    const int wave  = tid >> 5;          // 0..15
    const int lane  = tid & 31;
    const int m     = lane & 15;         // row-in-tile (A) / col-in-tile (B, C)
    const int khalf = (lane >> 4) << 1;  // 0 or 2: which K-pair this lane holds
    const int crow0 = (lane >> 4) << 3;  // C-fragment row base: 0 or 8

    // ---------------- FC1: one 16x16 tile of h per wave ----------------
    {
        const int mt = wave >> 2;        // batch-tile 0..3
        const int nt = wave & 3;         // squeeze-tile 0..3
        v8f c = {};
        for (int k = 0; k < CCH; k += 4) {
            const int kk = k + khalf;
            v2f a, b;
            a[0] = s[(mt * 16 + m) * CCH + kk];
            a[1] = s[(mt * 16 + m) * CCH + kk + 1];
            b[0] = w1[(nt * 16 + m) * CCH + kk];     // B[k][n] = w1[n][k]
            b[1] = w1[(nt * 16 + m) * CCH + kk + 1];
            c = __builtin_amdgcn_wmma_f32_16x16x4_f32(
                    false, a, false, b, (short)0, c, false, false);
        }
        const float bias = b1[nt * 16 + m];
        #pragma unroll
        for (int r = 0; r < 8; ++r) {
            float v = c[r] + bias;
            v = v > 0.0f ? v : 0.0f;                 // ReLU fused on writeback
            h_lds[(mt * 16 + crow0 + r) * HSTR + nt * 16 + m] = v;
        }
    }
    __syncthreads();

    // ---------------- FC2: 64 tiles of g, 4 per wave ----------------
    #pragma unroll
    for (int tt = 0; tt < 4; ++tt) {
        const int t  = wave * 4 + tt;
        const int mt = t >> 4;           // batch-tile 0..3
        const int nt = t & 15;           // channel-tile 0..15
        v8f c = {};
        for (int k = 0; k < SQC; k += 4) {
            const int kk = k + khalf;
            v2f a, b;
            a[0] = h_lds[(mt * 16 + m) * HSTR + kk];
            a[1] = h_lds[(mt * 16 + m) * HSTR + kk + 1];
            b[0] = w2[(nt * 16 + m) * SQC + kk];     // B[k][n] = w2[n][k]
            b[1] = w2[(nt * 16 + m) * SQC + kk + 1];
            c = __builtin_amdgcn_wmma_f32_16x16x4_f32(
                    false, a, false, b, (short)0, c, false, false);
        }
        const float bias = b2[nt * 16 + m];
        #pragma unroll
        for (int r = 0; r < 8; ++r) {
            float v = c[r] + bias;
            v = 1.0f / (1.0f + __expf(-v));          // sigmoid fused
            g[(mt * 16 + crow0 + r) * CCH + nt * 16 + m] = v;
        }
    }
}

// ---------------------------------------------------------------------------
// Kernel 3: broadcast scale. float4 (b128) in/out, grid divides exactly
// (12,845,056 float4 = 50176 blocks x 256 threads) so no tail branch.
// All 4 lanes of a float4 share one (b,c) since HW % 4 == 0.
// ---------------------------------------------------------------------------
__global__ void se_scale_kernel(const float* __restrict__ x,
                                const float* __restrict__ g,
                                float* __restrict__ out)
{
    const size_t i = (size_t)blockIdx.x * blockDim.x + threadIdx.x; // float4 idx
    const int row  = (int)(i / HW4);                                // = b*C + c
    const float scale = g[row];

    float4 v = reinterpret_cast<const float4*>(x)[i];
    v.x *= scale; v.y *= scale; v.z *= scale; v.w *= scale;
    reinterpret_cast<float4*>(out)[i] = v;
}

// ---------------------------------------------------------------------------
extern "C" void kernel_launch(void* const* d_in, const int* in_sizes, int n_in,
                              void* d_out, int out_size, void* d_ws, size_t ws_size,
                              hipStream_t stream) {
    const float* x  = (const float*)d_in[0];   // [64,256,56,56]
    const float* w1 = (const float*)d_in[1];   // [64,256]
    const float* b1 = (const float*)d_in[2];   // [64]
    const float* w2 = (const float*)d_in[3];   // [256,64]
    const float* b2 = (const float*)d_in[4];   // [256]
    float* out = (float*)d_out;

    float* s = (float*)d_ws;        // [64,256] squeeze result (64 KB)
    float* g = s + NROWS;           // [64,256] gate           (64 KB)

    // 1) pool: 16384 rows, 8 waves (rows) per 256-thread block -> 2048 blocks
    se_pool_kernel<<<NROWS / 8, 256, 0, stream>>>(x, s);

    // 2) excitation MLP: one 16-wave workgroup, WMMA f32
    se_excite_kernel<<<1, 512, 0, stream>>>(s, w1, b1, w2, b2, g);

    // 3) scale: 12,845,056 float4 elements, exact grid
    se_scale_kernel<<<(NROWS * HW4) / 256, 256, 0, stream>>>(x, g, out);
}